// NeuroplasticLlama_73632919322672
// MI455X (gfx1250) — compile-verified
//
#include <hip/hip_runtime.h>
#include <hip/hip_bf16.h>

typedef __attribute__((ext_vector_type(2))) float v2f;
typedef __attribute__((ext_vector_type(8))) float v8f;
typedef __attribute__((ext_vector_type(4))) unsigned int v4u;
typedef __attribute__((ext_vector_type(4))) int v4i;
typedef __attribute__((ext_vector_type(8))) int v8i;

#define HID 4096
#define BLKW 32
#define NBLK 128
#define RANK 4
#define TOPK 3
#define TOK_PER_WG 16
#define THREADS 512
#define WAVES 16

__global__ __launch_bounds__(THREADS)
void sca_fused_kernel(const float* __restrict__ x,
                      const int* __restrict__ task_ids,
                      const float* __restrict__ task_emb,
                      const float* __restrict__ Wp,
                      const float* __restrict__ bp,
                      const float* __restrict__ centers,
                      const float* __restrict__ A,
                      const float* __restrict__ Bm,
                      const float* __restrict__ scale_p,
                      float* __restrict__ out)
{
    extern __shared__ float smem[];
    float* xs     = smem;                          // 16*4096 floats (256 KB): x rows, later out rows
    float* embs   = xs + TOK_PER_WG * HID;         // 4096 floats
    float* cstage = embs + HID;                    // 16 waves * 256 floats (partial WMMA C)
    float* coords = cstage + WAVES * 256;          // 16*3
    float* gates  = coords + TOK_PER_WG * 3;       // 16*3
    int*   topi   = (int*)(gates + TOK_PER_WG * 3);// 16*3

    const int g    = blockIdx.x;
    const int tid  = threadIdx.x;
    const int lane = tid & 31;
    const int wave = tid >> 5;
    const long token0 = (long)g * TOK_PER_WG;
    const int  b      = (int)(token0 >> 12);       // S = 4096, 16 | S so one batch per WG
    const float* xrow0 = x + token0 * HID;

    // ---------------- Phase A: TDM async load of 16 x-rows into LDS ----------------
    if (tid == 0) {
        unsigned long long ga = (unsigned long long)(uintptr_t)xrow0;
        unsigned int lds_off  = (unsigned int)(uintptr_t)xs;  // low 32 bits = LDS byte offset
        // D# group 0: count=1 | lds_addr | global_addr | type=2
        v4u g0 = { 1u,
                   lds_off,
                   (unsigned int)(ga & 0xFFFFFFFFu),
                   (unsigned int)(((ga >> 32) & 0x1FFFFFFu) | (2u << 30)) };
        // D# group 1: data_size=4B; tensor_dim0=4096; tensor_dim1=16; tile_dim0=4096;
        //             tile_dim1=16; tensor_dim0_stride=4096 (elements)
        v8i g1 = { (int)(2u << 16),        // [17:16] data_size = 2 (4 bytes)
                   (int)(4096u << 16),     // [79:48] tensor_dim0 -> low half here
                   (int)(16u << 16),       // tensor_dim0 hi = 0 ; [111:80] tensor_dim1 low = 16
                   (int)(4096u << 16),     // tensor_dim1 hi = 0 ; [127:112] tile_dim0 = 4096
                   (int)16,                // [143:128] tile_dim1 = 16 ; tile_dim2 = 0
                   (int)4096,              // [207:160] tensor_dim0_stride low
                   0, 0 };                 // stride hi = 0 ; tensor_dim1_stride = 0
        v4i g2 = { 0, 0, 0, 0 };
        v4i g3 = { 0, 0, 0, 0 };
        v8i g4 = { 0, 0, 0, 0, 0, 0, 0, 0 };
        __builtin_amdgcn_tensor_load_to_lds(g0, g1, g2, g3, g4, 0);
    }

    // Task-embedding row with regular loads while TDM streams x
    {
        const float* erow = task_emb + (long)task_ids[b] * HID;
        for (int i = tid; i < HID; i += THREADS) embs[i] = erow[i];
    }

    __builtin_amdgcn_s_wait_tensorcnt(0);
    __syncthreads();

    // ---------------- Phase B: coords = h @ Wp via WMMA f32 16x16x4, K-split over waves ----------------
    {
        v8f c8 = {0.f,0.f,0.f,0.f,0.f,0.f,0.f,0.f};
        const int   mt    = lane & 15;          // token row (A-matrix M)
        const int   khalf = (lane >> 4) * 2;    // 0 (lanes 0-15) or 2 (lanes 16-31)
        const int   n     = lane & 15;          // output column (B-matrix N)
        const int   ncol  = (n < 3) ? n : 0;    // clamped column -> unconditional load
        const float nmask = (n < 3) ? 1.0f : 0.0f;
        const int   kbase = wave * (HID / WAVES);
        #pragma unroll 4
        for (int k = kbase; k < kbase + HID / WAVES; k += 4) {
            v2f a, bf;
            a.x = xs[mt * HID + k + khalf]     + embs[k + khalf];
            a.y = xs[mt * HID + k + khalf + 1] + embs[k + khalf + 1];
            bf.x = Wp[(k + khalf)     * 3 + ncol] * nmask;
            bf.y = Wp[(k + khalf + 1) * 3 + ncol] * nmask;
            c8 = __builtin_amdgcn_wmma_f32_16x16x4_f32(
                    false, a, false, bf, (short)0, c8, false, false);
        }
        #pragma unroll
        for (int v = 0; v < 8; ++v) cstage[wave * 256 + lane * 8 + v] = c8[v];
    }
    __syncthreads();

    // Reduce K-split partials: element (M=t, N=n) lives at lane = n + (t>=8?16:0), vgpr = t&7
    if (tid < TOK_PER_WG * 3) {
        const int t = tid / 3, n = tid % 3;
        const int cl = n + ((t >= 8) ? 16 : 0);
        const int v  = t & 7;
        float s = bp[n];
        #pragma unroll
        for (int w = 0; w < WAVES; ++w) s += cstage[w * 256 + cl * 8 + v];
        coords[t * 3 + n] = s;
    }
    __syncthreads();

    // ---------------- Phase C: routing (top-3 Gaussian score + softmax), one thread per token ----------------
    if (tid < TOK_PER_WG) {
        const float cx = coords[tid*3+0], cy = coords[tid*3+1], cz = coords[tid*3+2];
        float s0 = -1e30f, s1 = -1e30f, s2 = -1e30f;
        int   i0 = 0, i1 = 0, i2 = 0;
        for (int nb = 0; nb < NBLK; ++nb) {
            const float dx = cx - centers[nb*3+0];
            const float dy = cy - centers[nb*3+1];
            const float dz = cz - centers[nb*3+2];
            const float sc = -(dx*dx + dy*dy + dz*dz) * 0.5f;  // sigma = 1
            if (sc > s0)      { s2=s1; i2=i1; s1=s0; i1=i0; s0=sc; i0=nb; }
            else if (sc > s1) { s2=s1; i2=i1; s1=sc; i1=nb; }
            else if (sc > s2) { s2=sc; i2=nb; }
        }
        const float e0 = __expf(s0 - s0), e1 = __expf(s1 - s0), e2 = __expf(s2 - s0);
        const float inv = 1.0f / (e0 + e1 + e2);
        gates[tid*3+0] = e0 * inv; gates[tid*3+1] = e1 * inv; gates[tid*3+2] = e2 * inv;
        topi [tid*3+0] = i0;       topi [tid*3+1] = i1;       topi [tid*3+2] = i2;
    }
    __syncthreads();

    // ---------------- Phase D: sparse low-rank adapter, one wave per token ----------------
    {
        const int t = wave;
        int    blks[TOPK];
        float  gs[TOPK];
        float4 zv[TOPK];
        #pragma unroll
        for (int kk = 0; kk < TOPK; ++kk) { blks[kk] = topi[t*3+kk]; gs[kk] = gates[t*3+kk]; }

        // zv[kk] = h . A[blk]  (rank-4), coalesced float4 A loads, lane-interleaved over H
        #pragma unroll
        for (int kk = 0; kk < TOPK; ++kk) {
            const float4* Ab4 = (const float4*)(A + (long)blks[kk] * HID * RANK); // [HID] x float4
            float4 acc = make_float4(0.f, 0.f, 0.f, 0.f);
            #pragma unroll 4
            for (int i = lane; i < HID; i += 32) {
                const float  hv = xs[t * HID + i] + embs[i];
                const float4 av = Ab4[i];
                acc.x = fmaf(hv, av.x, acc.x);
                acc.y = fmaf(hv, av.y, acc.y);
                acc.z = fmaf(hv, av.z, acc.z);
                acc.w = fmaf(hv, av.w, acc.w);
            }
            #pragma unroll
            for (int m = 1; m < 32; m <<= 1) {
                acc.x += __shfl_xor(acc.x, m, 32);
                acc.y += __shfl_xor(acc.y, m, 32);
                acc.z += __shfl_xor(acc.z, m, 32);
                acc.w += __shfl_xor(acc.w, m, 32);
            }
            zv[kk] = acc;   // every lane holds the full rank-4 z vector
        }

        const float scale = *scale_p;
        // delta: lane = column c in the 32-wide block slice; top-k blocks are distinct per token
        #pragma unroll
        for (int kk = 0; kk < TOPK; ++kk) {
            const float* Bb = Bm + (long)blks[kk] * RANK * BLKW;
            const float d = zv[kk].x * Bb[0*BLKW + lane] + zv[kk].y * Bb[1*BLKW + lane]
                          + zv[kk].z * Bb[2*BLKW + lane] + zv[kk].w * Bb[3*BLKW + lane];
            xs[t * HID + blks[kk] * BLKW + lane] += scale * gs[kk] * d;  // out = x + scale*gate*delta
        }
    }
    __syncthreads();

    // ---------------- Phase E: stream the 16 finished rows to global ----------------
    {
        const float4* xs4  = (const float4*)xs;
        float4*       out4 = (float4*)(out + token0 * HID);
        for (int j = tid; j < TOK_PER_WG * HID / 4; j += THREADS) out4[j] = xs4[j];
    }
}

extern "C" void kernel_launch(void* const* d_in, const int* in_sizes, int n_in,
                              void* d_out, int out_size, void* d_ws, size_t ws_size,
                              hipStream_t stream) {
    const float* x        = (const float*)d_in[0];
    const int*   task_ids = (const int*)  d_in[1];
    const float* task_emb = (const float*)d_in[2];
    const float* Wp       = (const float*)d_in[3];
    const float* bp       = (const float*)d_in[4];
    const float* centers  = (const float*)d_in[5];
    const float* A        = (const float*)d_in[6];
    const float* Bm       = (const float*)d_in[7];
    const float* scale    = (const float*)d_in[8];
    float*       out      = (float*)d_out;

    const int ntok = in_sizes[0] / HID;        // B*S = 16384
    const int ngrp = ntok / TOK_PER_WG;        // 1024 workgroups

    const size_t shmem =
        (size_t)(TOK_PER_WG * HID      // xs
               + HID                   // embs
               + WAVES * 256           // cstage
               + TOK_PER_WG * 3        // coords
               + TOK_PER_WG * 3        // gates
               + TOK_PER_WG * 3)       // topi
        * sizeof(float);               // ~295 KB < 320 KB WGP LDS

    sca_fused_kernel<<<ngrp, THREADS, shmem, stream>>>(
        x, task_ids, task_emb, Wp, bp, centers, A, Bm, scale, out);
}